// molGCN_NNConv_41618233099024
// MI455X (gfx1250) — compile-verified
//
#include <hip/hip_runtime.h>
#include <hip/hip_bf16.h>

typedef __attribute__((ext_vector_type(2))) float v2f;
typedef __attribute__((ext_vector_type(8))) float v8f;

#define NNODE 60000
#define NEDGE 240000
#define CH    32          // NC = H1 = H2 = OC = 32
#define ECH   10
#define NG    2048
#define NCLS  12
#define KZ    (11*CH)     // 352: k = c*32 + i, c in [0,10] (c==10 -> bias term)
#define TE    32          // edges per wave tile-pair (two 16-row A tiles)
#define NT_E2 (NEDGE/TE)  // 7500 edge tile-pairs
#define NT_N  (NNODE/16)  // 3750 node tiles
#define NT_G  (NG/16)     // 128 graph tiles

__device__ __forceinline__ v8f wmma4(v2f a, v2f b, v8f c) {
  // V_WMMA_F32_16X16X4_F32 : D = A(16x4 f32) * B(4x16 f32) + C(16x16 f32)
  return __builtin_amdgcn_wmma_f32_16x16x4_f32(false, a, false, b, (short)0, c,
                                               false, false);
}

__device__ __forceinline__ void wave_lds_fence() {
  // cross-lane LDS visibility within a wave (LDS ops retire in order)
  asm volatile("s_wait_dscnt 0" ::: "memory");
}

// ---------------------------------------------------------------------------
// Repack (We, be) -> M[k][o], k = c*32 + i :
//   M[c*32+i][o] = (c < 10) ? We[(i*32+o)*10 + c] : be[i*32+o]
// so that msg_e = (ea'_e outer h_src) @ M, with ea'[10] = 1.
// ---------------------------------------------------------------------------
__global__ void pack_M_k(const float* __restrict__ We,
                         const float* __restrict__ be,
                         float* __restrict__ M) {
  int idx = blockIdx.x * blockDim.x + threadIdx.x;
  if (idx >= KZ * CH) return;
  int k = idx >> 5, o = idx & 31;
  int c = k >> 5,  i = k & 31;
  M[idx] = (c < ECH) ? We[(i * CH + o) * ECH + c] : be[i * CH + o];
}

// ---------------------------------------------------------------------------
// Root GEMM: out[n][:] = b + h[n][:] @ Wr.T      (N x 32) = (N x 32)(32 x 32)
// ---------------------------------------------------------------------------
__global__ __launch_bounds__(128) void root_k(const float* __restrict__ h,
                                              const float* __restrict__ Wr,
                                              const float* __restrict__ bias,
                                              float* __restrict__ out,
                                              int ntiles) {
  __shared__ float wr[CH * CH];     // wr[k*32+o] = Wr[o][k]  (B = Wr^T)
  __shared__ float bs[CH];
  __shared__ float hs[4][16 * 33];
  int tid = threadIdx.x;
  for (int idx = tid; idx < CH * CH; idx += 128) {
    int k = idx >> 5, o = idx & 31;
    wr[idx] = Wr[o * CH + k];
  }
  if (tid < CH) bs[tid] = bias[tid];
  __syncthreads();

  int wid = tid >> 5, lane = tid & 31;
  int m = lane & 15, hi = lane >> 4;
  int tile = blockIdx.x * 4 + wid;
  if (tile >= ntiles) return;
  int base = tile * 16;
  float* hsw = hs[wid];
  for (int r = 0; r < 16; ++r)
    hsw[r * 33 + lane] = h[(base + r) * CH + lane];
  wave_lds_fence();

  v8f acc0 = {}, acc1 = {};
#pragma unroll
  for (int t = 0; t < 8; ++t) {
    int kk = 4 * t + (hi ? 2 : 0);
    v2f a  = { hsw[m * 33 + kk], hsw[m * 33 + kk + 1] };
    v2f b0 = { wr[kk * CH + m],        wr[(kk + 1) * CH + m] };
    v2f b1 = { wr[kk * CH + m + 16],   wr[(kk + 1) * CH + m + 16] };
    acc0 = wmma4(a, b0, acc0);
    acc1 = wmma4(a, b1, acc1);
  }
#pragma unroll
  for (int r = 0; r < 8; ++r) {
    int mr = r + (hi ? 8 : 0);
    out[(base + mr) * CH + m]      = acc0[r] + bs[m];
    out[(base + mr) * CH + m + 16] = acc1[r] + bs[m + 16];
  }
}

// ---------------------------------------------------------------------------
// Edge GEMM + scatter, register-blocked: each wave owns 32 edges (two 16-row
// A-tiles) x full 32 output columns (two 16-col N-tiles). Each B fragment
// read from LDS feeds TWO wmma ops -> 6 ds loads per 4 wmma per k-step.
// M (352x32, 45 KB) cached in LDS per block; per-wave private h/ea/dst tiles.
// ---------------------------------------------------------------------------
__global__ __launch_bounds__(96) void edge_k(const float* __restrict__ h,
                                             const int* __restrict__ src,
                                             const int* __restrict__ dst,
                                             const float* __restrict__ ea,
                                             const float* __restrict__ M,
                                             float* __restrict__ out) {
  __shared__ float Ml[KZ * CH];     // 45056 B
  __shared__ float hs[3][TE * 33];  // 3 * 4224 B
  __shared__ float eal[3][TE * 12]; // 3 * 1536 B
  __shared__ int   dl[3][TE];       // 3 * 128 B   (total 62720 B)
  int tid = threadIdx.x;
  for (int idx = tid; idx < KZ * CH; idx += 96) Ml[idx] = M[idx];
  __syncthreads();

  int wid = tid >> 5, lane = tid & 31;
  int m = lane & 15, hi = lane >> 4;
  float* hsw = hs[wid];
  float* eaw = eal[wid];
  int*   dw  = dl[wid];
  int stride = gridDim.x * 3;

  for (int tile = blockIdx.x * 3 + wid; tile < NT_E2; tile += stride) {
    int base = tile * TE;
    dw[lane] = dst[base + lane];
    eaw[lane * 12 + 10] = 1.0f;              // bias lane of ea'
    eaw[lane * 12 + 11] = 0.0f;
    for (int t = lane; t < TE * ECH; t += 32) {
      int e = t / ECH, c = t - e * ECH;
      eaw[e * 12 + c] = ea[(base + e) * ECH + c];
    }
    for (int r = 0; r < TE; ++r) {           // gather source features
      int s = src[base + r];                 // uniform -> scalarized
      hsw[r * 33 + lane] = h[s * CH + lane];
    }
    wave_lds_fence();

    v8f acc0 = {}, acc1 = {}, acc2 = {}, acc3 = {};
#pragma unroll 4
    for (int t = 0; t < KZ / 4; ++t) {       // 88 k-steps
      int kk = 4 * t + (hi ? 2 : 0);
      int c = kk >> 5, i = kk & 31;          // i <= 30, so i+1 same c
      float ev0 = eaw[m * 12 + c];
      float ev1 = eaw[(m + 16) * 12 + c];
      v2f a0 = { hsw[m * 33 + i] * ev0,        hsw[m * 33 + i + 1] * ev0 };
      v2f a1 = { hsw[(m + 16) * 33 + i] * ev1, hsw[(m + 16) * 33 + i + 1] * ev1 };
      v2f b0 = { Ml[kk * CH + m],      Ml[(kk + 1) * CH + m] };
      v2f b1 = { Ml[kk * CH + m + 16], Ml[(kk + 1) * CH + m + 16] };
      acc0 = wmma4(a0, b0, acc0);
      acc1 = wmma4(a0, b1, acc1);
      acc2 = wmma4(a1, b0, acc2);
      acc3 = wmma4(a1, b1, acc3);
    }
#pragma unroll
    for (int r = 0; r < 8; ++r) {
      int mr = r + (hi ? 8 : 0);
      int d0 = dw[mr];
      int d1 = dw[mr + 16];
      unsafeAtomicAdd(&out[d0 * CH + m],      acc0[r]);
      unsafeAtomicAdd(&out[d0 * CH + m + 16], acc1[r]);
      unsafeAtomicAdd(&out[d1 * CH + m],      acc2[r]);
      unsafeAtomicAdd(&out[d1 * CH + m + 16], acc3[r]);
    }
  }
}

__global__ void relu_k(float* __restrict__ x, int n) {
  int i = blockIdx.x * blockDim.x + threadIdx.x;
  if (i < n) x[i] = fmaxf(x[i], 0.0f);
}

__global__ void zero_k(float* __restrict__ x, int n) {
  int i = blockIdx.x * blockDim.x + threadIdx.x;
  if (i < n) x[i] = 0.0f;
}

__global__ void pool_k(const float* __restrict__ h,
                       const int* __restrict__ batch,
                       float* __restrict__ pooled) {
  int idx = blockIdx.x * blockDim.x + threadIdx.x;
  if (idx >= NNODE * CH) return;
  int n = idx >> 5, o = idx & 31;
  unsafeAtomicAdd(&pooled[batch[n] * CH + o], h[idx]);
}

// ---------------------------------------------------------------------------
// Final linear: out = pooled(G x 32) @ Wlin.T(32 x 12) + blin, via WMMA with
// columns >= 12 zero-padded in B and masked on store.
// ---------------------------------------------------------------------------
__global__ __launch_bounds__(128) void final_k(const float* __restrict__ pooled,
                                               const float* __restrict__ Wlin,
                                               const float* __restrict__ blin,
                                               float* __restrict__ out) {
  __shared__ float wl[CH * 16];     // wl[k*16+n] = (n<12) ? Wlin[n][k] : 0
  __shared__ float bl[16];
  __shared__ float hs[4][16 * 33];
  int tid = threadIdx.x;
  for (int idx = tid; idx < CH * 16; idx += 128) {
    int k = idx >> 4, n = idx & 15;
    wl[idx] = (n < NCLS) ? Wlin[n * CH + k] : 0.0f;
  }
  if (tid < 16) bl[tid] = (tid < NCLS) ? blin[tid] : 0.0f;
  __syncthreads();

  int wid = tid >> 5, lane = tid & 31;
  int m = lane & 15, hi = lane >> 4;
  int tile = blockIdx.x * 4 + wid;
  if (tile >= NT_G) return;
  int base = tile * 16;
  float* hsw = hs[wid];
  for (int r = 0; r < 16; ++r)
    hsw[r * 33 + lane] = pooled[(base + r) * CH + lane];
  wave_lds_fence();

  v8f acc = {};
#pragma unroll
  for (int t = 0; t < 8; ++t) {
    int kk = 4 * t + (hi ? 2 : 0);
    v2f a = { hsw[m * 33 + kk], hsw[m * 33 + kk + 1] };
    v2f b = { wl[kk * 16 + m], wl[(kk + 1) * 16 + m] };
    acc = wmma4(a, b, acc);
  }
  if (m < NCLS) {
#pragma unroll
    for (int r = 0; r < 8; ++r) {
      int mr = r + (hi ? 8 : 0);
      out[(base + mr) * NCLS + m] = acc[r] + bl[m];
    }
  }
}

// ---------------------------------------------------------------------------
extern "C" void kernel_launch(void* const* d_in, const int* in_sizes, int n_in,
                              void* d_out, int out_size, void* d_ws, size_t ws_size,
                              hipStream_t stream) {
  const float* x    = (const float*)d_in[0];
  const int*   ei   = (const int*)d_in[1];
  const float* ea   = (const float*)d_in[2];
  const int*   bat  = (const int*)d_in[3];
  const float* We1  = (const float*)d_in[4];
  const float* be1  = (const float*)d_in[5];
  const float* Wr1  = (const float*)d_in[6];
  const float* b1   = (const float*)d_in[7];
  const float* We2  = (const float*)d_in[8];
  const float* be2  = (const float*)d_in[9];
  const float* Wr2  = (const float*)d_in[10];
  const float* b2   = (const float*)d_in[11];
  const float* We3  = (const float*)d_in[12];
  const float* be3  = (const float*)d_in[13];
  const float* Wr3  = (const float*)d_in[14];
  const float* b3   = (const float*)d_in[15];
  const float* Wlin = (const float*)d_in[16];
  const float* blin = (const float*)d_in[17];
  const int* srcp = ei;
  const int* dstp = ei + NEDGE;

  float* ws = (float*)d_ws;
  float* hA = ws;                       // N*32
  float* hB = hA + NNODE * CH;          // N*32
  float* M1 = hB + NNODE * CH;          // 352*32
  float* M2 = M1 + KZ * CH;
  float* M3 = M2 + KZ * CH;
  float* pooled = M3 + KZ * CH;         // G*32

  const int EB = 256;
  int gM = (KZ * CH + EB - 1) / EB;
  pack_M_k<<<gM, EB, 0, stream>>>(We1, be1, M1);
  pack_M_k<<<gM, EB, 0, stream>>>(We2, be2, M2);
  pack_M_k<<<gM, EB, 0, stream>>>(We3, be3, M3);

  int gRoot = (NT_N + 3) / 4;
  int gElem = (NNODE * CH + EB - 1) / EB;

  // layer 1: x -> hA
  root_k<<<gRoot, 128, 0, stream>>>(x, Wr1, b1, hA, NT_N);
  edge_k<<<512, 96, 0, stream>>>(x, srcp, dstp, ea, M1, hA);
  relu_k<<<gElem, EB, 0, stream>>>(hA, NNODE * CH);
  // layer 2: hA -> hB
  root_k<<<gRoot, 128, 0, stream>>>(hA, Wr2, b2, hB, NT_N);
  edge_k<<<512, 96, 0, stream>>>(hA, srcp, dstp, ea, M2, hB);
  relu_k<<<gElem, EB, 0, stream>>>(hB, NNODE * CH);
  // layer 3: hB -> hA
  root_k<<<gRoot, 128, 0, stream>>>(hB, Wr3, b3, hA, NT_N);
  edge_k<<<512, 96, 0, stream>>>(hB, srcp, dstp, ea, M3, hA);
  relu_k<<<gElem, EB, 0, stream>>>(hA, NNODE * CH);

  // pool + classify
  zero_k<<<(NG * CH + EB - 1) / EB, EB, 0, stream>>>(pooled, NG * CH);
  pool_k<<<gElem, EB, 0, stream>>>(hA, bat, pooled);
  final_k<<<NT_G / 4, 128, 0, stream>>>(pooled, Wlin, blin, (float*)d_out);
}